// Net_39453569581387
// MI455X (gfx1250) — compile-verified
//
#include <hip/hip_runtime.h>
#include <hip/hip_bf16.h>
#include <float.h>

// ---------------------------------------------------------------------------
// DGCNN forward on gfx1250 (MI455X).  All GEMMs via V_WMMA_F32_16X16X4_F32.
// ---------------------------------------------------------------------------

#define NGRAPH 16
#define NPERG  2048
#define NTOT   (NGRAPH * NPERG)
#define KNN    20
#define NCLS   10

typedef __attribute__((ext_vector_type(2))) float v2f;
typedef __attribute__((ext_vector_type(8))) float v8f;

// D = A(16x4, f32) * B(4x16, f32) + C(16x16, f32)   -- exact fp32 matrix op
__device__ __forceinline__ v8f wmma4(v2f a, v2f b, v8f c) {
  return __builtin_amdgcn_wmma_f32_16x16x4_f32(
      /*neg_a=*/false, a, /*neg_b=*/false, b,
      /*c_mod=*/(short)0, c, /*reuse_a=*/false, /*reuse_b=*/false);
}

// order-preserving float <-> uint mapping (for atomicMax-based max pooling)
__device__ __forceinline__ unsigned f2ord(float f) {
  unsigned u = __float_as_uint(f);
  return u ^ ((unsigned)((int)u >> 31) | 0x80000000u);
}
__device__ __forceinline__ float ord2f(unsigned u) {
  return __uint_as_float((u & 0x80000000u) ? (u ^ 0x80000000u) : ~u);
}

// ---------------------------------------------------------------------------
// x0 = concat(pos, x)  -> [N,4]
// ---------------------------------------------------------------------------
__global__ void build_x0_kernel(const float* __restrict__ pos,
                                const float* __restrict__ xf,
                                float* __restrict__ x0) {
  int i = blockIdx.x * blockDim.x + threadIdx.x;
  if (i < NTOT) {
    x0[i * 4 + 0] = pos[i * 3 + 0];
    x0[i * 4 + 1] = pos[i * 3 + 1];
    x0[i * 4 + 2] = pos[i * 3 + 2];
    x0[i * 4 + 3] = xf[i];
  }
}

// ---------------------------------------------------------------------------
// Per-graph kNN (K=20, self excluded).  256 queries/block, candidate table
// staged in LDS (2048 x float4 = 32 KB), register insertion-sort top-K.
// Neighbor order within K is irrelevant downstream (max-aggregation).
// ---------------------------------------------------------------------------
__global__ __launch_bounds__(256) void knn_kernel(const float* __restrict__ x0,
                                                  int* __restrict__ nbr) {
  __shared__ float4 sF[NPERG];
  const int g    = blockIdx.x / (NPERG / 256);
  const int q0   = (blockIdx.x % (NPERG / 256)) * 256;
  const int base = g * NPERG;
  for (int i = threadIdx.x; i < NPERG; i += 256)
    sF[i] = *(const float4*)(x0 + (size_t)(base + i) * 4);
  __syncthreads();

  const int q = q0 + threadIdx.x;
  const float4 me = sF[q];

  float bd[KNN];
  int   bi[KNN];
#pragma unroll
  for (int k = 0; k < KNN; ++k) { bd[k] = FLT_MAX; bi[k] = 0; }

  for (int j = 0; j < NPERG; ++j) {
    if (j == q) continue;
    float4 o = sF[j];
    float dx = me.x - o.x, dy = me.y - o.y, dz = me.z - o.z, dw = me.w - o.w;
    float d = dx * dx + dy * dy + dz * dz + dw * dw;
    if (d < bd[KNN - 1]) {
      float cd = d; int ci = j;
#pragma unroll
      for (int k = 0; k < KNN; ++k) {
        if (cd < bd[k]) {
          float td = bd[k]; bd[k] = cd; cd = td;
          int   ti = bi[k]; bi[k] = ci; ci = ti;
        }
      }
    }
  }
#pragma unroll
  for (int k = 0; k < KNN; ++k)
    nbr[(size_t)(base + q) * KNN + k] = base + bi[k];
}

// ---------------------------------------------------------------------------
// Fused EdgeConv: msg = MLP([x_i, x_j - x_i]), max over K neighbors.
// One wave handles 16 nodes.  Neighbor slot t forms a 16x(2*FIN) A tile
// (row m = node m's t-th neighbor), so max over K is an elementwise max of
// C tiles in identical layout.  Layers: 2*FIN -> FOUT (-> FOUT -> FOUT).
// A fragments read from padded LDS; B fragments read from global (L2-hot).
// ---------------------------------------------------------------------------
template <int FIN, int FOUT, int NL>
__global__ __launch_bounds__(32) void edgeconv_kernel(
    const float* __restrict__ xin, const int* __restrict__ nbr,
    const float* __restrict__ W0, const float* __restrict__ B0,
    const float* __restrict__ W1, const float* __restrict__ B1,
    const float* __restrict__ W2, const float* __restrict__ B2,
    float* __restrict__ xout) {
  constexpr int FMSG = 2 * FIN;
  constexpr int AMAX = (FMSG > FOUT) ? FMSG : FOUT;
  constexpr int AST  = AMAX + 4;  // pad: conflict-free A-fragment reads
  constexpr int NT   = FOUT / 16;

  __shared__ float sA[16 * AST];
  __shared__ int   sJ[16];

  const int lane  = threadIdx.x;
  const int nb    = blockIdx.x * 16;
  const int lrow  = lane & 15;
  const int khalf = (lane >> 4) << 1;      // 0 (lanes 0-15) or 2 (lanes 16-31)
  const int rhalf = (lane >> 4) << 3;      // C/D row offset: 0 or 8

  if (NL == 1) {  // x_i half persists across all K tiles
    for (int e = lane; e < 16 * FIN; e += 32) {
      int m = e / FIN, c = e % FIN;
      sA[m * AST + c] = xin[(size_t)(nb + m) * FIN + c];
    }
  }

  v8f mx[NT];
#pragma unroll
  for (int nt = 0; nt < NT; ++nt)
#pragma unroll
    for (int v = 0; v < 8; ++v) mx[nt][v] = -FLT_MAX;

  for (int t = 0; t < KNN; ++t) {
    if (lane < 16) sJ[lane] = nbr[(size_t)(nb + lane) * KNN + t];
    __syncthreads();
    if (NL > 1) {  // hidden restaging clobbered x_i half last tile: rebuild
      for (int e = lane; e < 16 * FIN; e += 32) {
        int m = e / FIN, c = e % FIN;
        sA[m * AST + c] = xin[(size_t)(nb + m) * FIN + c];
      }
    }
    for (int e = lane; e < 16 * FIN; e += 32) {
      int m = e / FIN, c = e % FIN;
      sA[m * AST + FIN + c] =
          xin[(size_t)sJ[m] * FIN + c] - xin[(size_t)(nb + m) * FIN + c];
    }
    __syncthreads();

    int kdim = FMSG;
    v8f acc[NT];
    for (int L = 0; L < NL; ++L) {
      const float* W  = (L == 0) ? W0 : ((L == 1) ? W1 : W2);
      const float* Bb = (L == 0) ? B0 : ((L == 1) ? B1 : B2);
#pragma unroll
      for (int nt = 0; nt < NT; ++nt)
#pragma unroll
        for (int v = 0; v < 8; ++v) acc[nt][v] = 0.f;

      for (int s = 0; s < kdim / 4; ++s) {
        const int ac = 4 * s + khalf;
        v2f a;
        a.x = sA[lrow * AST + ac];
        a.y = sA[lrow * AST + ac + 1];
#pragma unroll
        for (int nt = 0; nt < NT; ++nt) {
          const int col = nt * 16 + lrow;
          v2f b;
          b.x = W[(size_t)ac * FOUT + col];
          b.y = W[(size_t)(ac + 1) * FOUT + col];
          acc[nt] = wmma4(a, b, acc[nt]);
        }
      }

      if (L < NL - 1) {  // bias + ReLU, restage C-layout -> A-layout in LDS
        __syncthreads();
#pragma unroll
        for (int nt = 0; nt < NT; ++nt) {
          float bia = Bb[nt * 16 + lrow];
#pragma unroll
          for (int v = 0; v < 8; ++v) {
            float val = fmaxf(acc[nt][v] + bia, 0.f);
            sA[(v + rhalf) * AST + nt * 16 + lrow] = val;
          }
        }
        __syncthreads();
        kdim = FOUT;
      } else {  // final layer: bias then running max over neighbor slots
#pragma unroll
        for (int nt = 0; nt < NT; ++nt) {
          float bia = Bb[nt * 16 + lrow];
#pragma unroll
          for (int v = 0; v < 8; ++v)
            mx[nt][v] = fmaxf(mx[nt][v], acc[nt][v] + bia);
        }
      }
    }
    __syncthreads();
  }

#pragma unroll
  for (int nt = 0; nt < NT; ++nt)
#pragma unroll
    for (int v = 0; v < 8; ++v)
      xout[(size_t)(nb + v + rhalf) * FOUT + nt * 16 + lrow] = mx[nt][v];
}

// ---------------------------------------------------------------------------
__global__ void init_out2_kernel(unsigned* __restrict__ o) {
  int i = blockIdx.x * blockDim.x + threadIdx.x;
  if (i < NGRAPH * 1024) o[i] = 0u;  // ordered-uint lower bound
}

// ---------------------------------------------------------------------------
// out1 = concat(x1,x2,x3) @ l1w, fused with global max-pool over nodes.
// Block = 4 waves, stages one 16x256 A tile; wave w computes 64 columns of
// the 256-column group blockIdx.y; per-tile node max -> atomicMax (ordered
// uint).  l1b folded in by the head kernel (constant shift commutes w/ max).
// ---------------------------------------------------------------------------
__global__ __launch_bounds__(128) void lin1_maxpool_kernel(
    const float* __restrict__ x1, const float* __restrict__ x2,
    const float* __restrict__ x3, const float* __restrict__ l1w,
    unsigned* __restrict__ out2) {
  constexpr int AST = 260;
  __shared__ float sA[16 * AST];
  const int tid   = threadIdx.x;
  const int lane  = tid & 31;
  const int wave  = tid >> 5;
  const int nb    = blockIdx.x * 16;
  const int g     = nb / NPERG;
  const int cbase = blockIdx.y * 256;

  for (int e = tid; e < 16 * 256; e += 128) {
    int m = e >> 8, c = e & 255;
    float v;
    if (c < 64)       v = x1[(size_t)(nb + m) * 64  + c];
    else if (c < 128) v = x2[(size_t)(nb + m) * 64  + (c - 64)];
    else              v = x3[(size_t)(nb + m) * 128 + (c - 128)];
    sA[m * AST + c] = v;
  }
  __syncthreads();

  const int lrow  = lane & 15;
  const int khalf = (lane >> 4) << 1;

  v8f acc[4];
#pragma unroll
  for (int nt = 0; nt < 4; ++nt)
#pragma unroll
    for (int v = 0; v < 8; ++v) acc[nt][v] = 0.f;

  for (int s = 0; s < 64; ++s) {
    const int ac = 4 * s + khalf;
    v2f a;
    a.x = sA[lrow * AST + ac];
    a.y = sA[lrow * AST + ac + 1];
#pragma unroll
    for (int nt = 0; nt < 4; ++nt) {
      const int col = cbase + wave * 64 + nt * 16 + lrow;
      v2f b;
      b.x = l1w[(size_t)ac * 1024 + col];
      b.y = l1w[(size_t)(ac + 1) * 1024 + col];
      acc[nt] = wmma4(a, b, acc[nt]);
    }
  }

#pragma unroll
  for (int nt = 0; nt < 4; ++nt) {
    float m8 = acc[nt][0];
#pragma unroll
    for (int v = 1; v < 8; ++v) m8 = fmaxf(m8, acc[nt][v]);
    float m16 = fmaxf(m8, __shfl_xor(m8, 16, 32));  // rows 0-7 vs 8-15
    if (lane < 16) {
      const int col = cbase + wave * 64 + nt * 16 + lrow;
      atomicMax(&out2[g * 1024 + col], f2ord(m16));
    }
  }
}

// ---------------------------------------------------------------------------
// Classifier head: (+l1b) -> 1024->512 relu -> 512->256 relu -> 256->10
// -> log_softmax.  One block per graph; trivially small.
// ---------------------------------------------------------------------------
__global__ __launch_bounds__(256) void head_kernel(
    const unsigned* __restrict__ out2, const float* __restrict__ l1b,
    const float* __restrict__ m1w, const float* __restrict__ m1b,
    const float* __restrict__ m2w, const float* __restrict__ m2b,
    const float* __restrict__ m3w, const float* __restrict__ m3b,
    float* __restrict__ out) {
  __shared__ float v[1024];
  __shared__ float h1[512];
  __shared__ float h2[256];
  __shared__ float h3[NCLS];
  __shared__ float lse_s;
  const int g = blockIdx.x, tid = threadIdx.x;

  for (int c = tid; c < 1024; c += 256) v[c] = ord2f(out2[g * 1024 + c]) + l1b[c];
  __syncthreads();
  for (int j = tid; j < 512; j += 256) {
    float s = m1b[j];
    for (int c = 0; c < 1024; ++c) s += v[c] * m1w[(size_t)c * 512 + j];
    h1[j] = fmaxf(s, 0.f);
  }
  __syncthreads();
  for (int j = tid; j < 256; j += 256) {
    float s = m2b[j];
    for (int c = 0; c < 512; ++c) s += h1[c] * m2w[(size_t)c * 256 + j];
    h2[j] = fmaxf(s, 0.f);
  }
  __syncthreads();
  if (tid < NCLS) {
    float s = m3b[tid];
    for (int c = 0; c < 256; ++c) s += h2[c] * m3w[(size_t)c * NCLS + tid];
    h3[tid] = s;
  }
  __syncthreads();
  if (tid == 0) {
    float mx = h3[0];
    for (int k = 1; k < NCLS; ++k) mx = fmaxf(mx, h3[k]);
    float s = 0.f;
    for (int k = 0; k < NCLS; ++k) s += expf(h3[k] - mx);
    lse_s = mx + logf(s);
  }
  __syncthreads();
  if (tid < NCLS) out[g * NCLS + tid] = h3[tid] - lse_s;
}

// ---------------------------------------------------------------------------
extern "C" void kernel_launch(void* const* d_in, const int* in_sizes, int n_in,
                              void* d_out, int out_size, void* d_ws,
                              size_t ws_size, hipStream_t stream) {
  const float* pos  = (const float*)d_in[0];
  const float* xf   = (const float*)d_in[1];
  /* d_in[2] = batch (layout known, unused) */
  const float* c1w1 = (const float*)d_in[3];  const float* c1b1 = (const float*)d_in[4];
  const float* c1w2 = (const float*)d_in[5];  const float* c1b2 = (const float*)d_in[6];
  const float* c1w3 = (const float*)d_in[7];  const float* c1b3 = (const float*)d_in[8];
  const float* c2w  = (const float*)d_in[9];  const float* c2b  = (const float*)d_in[10];
  const float* c3w  = (const float*)d_in[11]; const float* c3b  = (const float*)d_in[12];
  const float* l1w  = (const float*)d_in[13]; const float* l1b  = (const float*)d_in[14];
  const float* m1w  = (const float*)d_in[15]; const float* m1b  = (const float*)d_in[16];
  const float* m2w  = (const float*)d_in[17]; const float* m2b  = (const float*)d_in[18];
  const float* m3w  = (const float*)d_in[19]; const float* m3b  = (const float*)d_in[20];

  // workspace layout (~37 MB total)
  float*    x0  = (float*)d_ws;                          // N*4
  int*      nbr = (int*)(x0 + (size_t)NTOT * 4);         // N*20
  float*    x1v = (float*)(nbr + (size_t)NTOT * KNN);    // N*64
  float*    x2v = x1v + (size_t)NTOT * 64;               // N*64
  float*    x3v = x2v + (size_t)NTOT * 64;               // N*128
  unsigned* o2  = (unsigned*)(x3v + (size_t)NTOT * 128); // 16*1024

  build_x0_kernel<<<NTOT / 256, 256, 0, stream>>>(pos, xf, x0);
  knn_kernel<<<NTOT / 256, 256, 0, stream>>>(x0, nbr);

  edgeconv_kernel<4, 64, 3><<<NTOT / 16, 32, 0, stream>>>(
      x0, nbr, c1w1, c1b1, c1w2, c1b2, c1w3, c1b3, x1v);
  edgeconv_kernel<64, 64, 1><<<NTOT / 16, 32, 0, stream>>>(
      x1v, nbr, c2w, c2b, nullptr, nullptr, nullptr, nullptr, x2v);
  edgeconv_kernel<64, 128, 1><<<NTOT / 16, 32, 0, stream>>>(
      x2v, nbr, c3w, c3b, nullptr, nullptr, nullptr, nullptr, x3v);

  init_out2_kernel<<<(NGRAPH * 1024) / 256, 256, 0, stream>>>(o2);
  dim3 grid_lin(NTOT / 16, 4);
  lin1_maxpool_kernel<<<grid_lin, 128, 0, stream>>>(x1v, x2v, x3v, l1w, o2);

  head_kernel<<<NGRAPH, 256, 0, stream>>>(o2, l1b, m1w, m1b, m2w, m2b, m3w,
                                          m3b, (float*)d_out);
}